// RelMultiHeadAttention_10651518894752
// MI455X (gfx1250) — compile-verified
//
#include <hip/hip_runtime.h>
#include <math.h>

typedef __attribute__((ext_vector_type(2))) float v2f;
typedef __attribute__((ext_vector_type(8))) float v8f;

static constexpr int kS  = 2048;  // sequence length
static constexpr int kD  = 512;   // model dim
static constexpr int kH  = 8;     // heads
static constexpr int kDH = 64;    // head dim

__device__ __forceinline__ v8f wmma_f32(v2f a, v2f b, v8f c) {
  // V_WMMA_F32_16X16X4_F32 : D(16x16 f32) = A(16x4) * B(4x16) + C
  return __builtin_amdgcn_wmma_f32_16x16x4_f32(
      /*neg_a=*/false, a, /*neg_b=*/false, b,
      /*c_mod=*/(short)0, c, /*reuse_a=*/false, /*reuse_b=*/false);
}

// ---------------------------------------------------------------------------
// C = A[M,K] @ B[N,K]^T  (torch Linear), output scattered to head-major
// out[h][m][d] with h = n/64, d = n%64.  K = N = 512, M = 2048.
// Block = 256 threads = 8 waves; wave w owns columns [w*64, w*64+64).
// ---------------------------------------------------------------------------
__global__ void proj_gemm_head(const float* __restrict__ A,
                               const float* __restrict__ B,
                               float* __restrict__ out) {
  const int lane = threadIdx.x & 31;
  const int wave = threadIdx.x >> 5;
  const int half = lane >> 4;        // 0 or 1
  const int lr   = lane & 15;
  const int m0   = blockIdx.x * 16;
  const int n0   = wave * 64;

  const float* ap = A + (size_t)(m0 + lr) * kD + 2 * half;
  const float* bp = B + (size_t)(n0 + lr) * kD + 2 * half;

  v8f acc[4] = {v8f{}, v8f{}, v8f{}, v8f{}};
  for (int k = 0; k < kD; k += 4) {
    v2f a = *(const v2f*)(ap + k);
#pragma unroll
    for (int t = 0; t < 4; ++t) {
      v2f b = *(const v2f*)(bp + (size_t)t * 16 * kD + k);
      acc[t] = wmma_f32(a, b, acc[t]);
    }
  }
#pragma unroll
  for (int t = 0; t < 4; ++t) {
    int n = n0 + t * 16 + lr;
    int h = n >> 6, d = n & 63;
    float* op = out + (size_t)h * kS * kDH + d;
#pragma unroll
    for (int r = 0; r < 8; ++r) {
      int m = m0 + r + 8 * half;
      op[(size_t)m * kDH] = acc[t][r];
    }
  }
}

// ---------------------------------------------------------------------------
// Per head: BD[q,c] = sum_d (Q[h][q][d] + vb[h][d]) * P[h][c][d]
// Block = 8 waves covering 512 columns; grid = (S/16, S/512).
// ---------------------------------------------------------------------------
__global__ void bd_gemm(const float* __restrict__ Qh, const float* __restrict__ Ph,
                        const float* __restrict__ vb, float* __restrict__ BD, int h) {
  const int lane = threadIdx.x & 31;
  const int wave = threadIdx.x >> 5;
  const int half = lane >> 4;
  const int lr   = lane & 15;
  const int m0   = blockIdx.x * 16;
  const int n0   = blockIdx.y * 512 + wave * 64;

  const float* qp = Qh + (size_t)h * kS * kDH + (size_t)(m0 + lr) * kDH + 2 * half;
  const float* pp = Ph + (size_t)h * kS * kDH + (size_t)(n0 + lr) * kDH + 2 * half;
  const float* bb = vb + h * kDH + 2 * half;

  v8f acc[4] = {v8f{}, v8f{}, v8f{}, v8f{}};
  for (int k = 0; k < kDH; k += 4) {
    v2f a = *(const v2f*)(qp + k);
    v2f bias = *(const v2f*)(bb + k);
    a = a + bias;
#pragma unroll
    for (int t = 0; t < 4; ++t) {
      v2f b = *(const v2f*)(pp + (size_t)t * 16 * kDH + k);
      acc[t] = wmma_f32(a, b, acc[t]);
    }
  }
#pragma unroll
  for (int t = 0; t < 4; ++t) {
    int n = n0 + t * 16 + lr;
#pragma unroll
    for (int r = 0; r < 8; ++r) {
      int m = m0 + r + 8 * half;
      BD[(size_t)m * kS + n] = acc[t][r];
    }
  }
}

// ---------------------------------------------------------------------------
// Per head: SC[q,k] = ( (Q+u)·K^T  +  rel_shift(BD) ) * scale
// rel_shift (derived from the pad/reshape trick, verified on 3x3):
//   k <= q   : BD[q,  S-1-q+k]
//   k == q+1 : 0
//   k >  q+1 : BD[q+1, k-q-2]
// ---------------------------------------------------------------------------
__global__ void score_gemm(const float* __restrict__ Qh, const float* __restrict__ Kh,
                           const float* __restrict__ ub, const float* __restrict__ BD,
                           float* __restrict__ SC, int h, float scale) {
  const int lane = threadIdx.x & 31;
  const int wave = threadIdx.x >> 5;
  const int half = lane >> 4;
  const int lr   = lane & 15;
  const int m0   = blockIdx.x * 16;
  const int n0   = blockIdx.y * 512 + wave * 64;

  const float* qp = Qh + (size_t)h * kS * kDH + (size_t)(m0 + lr) * kDH + 2 * half;
  const float* kp = Kh + (size_t)h * kS * kDH + (size_t)(n0 + lr) * kDH + 2 * half;
  const float* bb = ub + h * kDH + 2 * half;

  v8f acc[4] = {v8f{}, v8f{}, v8f{}, v8f{}};
  for (int k = 0; k < kDH; k += 4) {
    v2f a = *(const v2f*)(qp + k);
    v2f bias = *(const v2f*)(bb + k);
    a = a + bias;
#pragma unroll
    for (int t = 0; t < 4; ++t) {
      v2f b = *(const v2f*)(kp + (size_t)t * 16 * kDH + k);
      acc[t] = wmma_f32(a, b, acc[t]);
    }
  }
#pragma unroll
  for (int t = 0; t < 4; ++t) {
    int n = n0 + t * 16 + lr;
#pragma unroll
    for (int r = 0; r < 8; ++r) {
      int m = m0 + r + 8 * half;
      float bd;
      if (n <= m)          bd = BD[(size_t)m * kS + (kS - 1 - m + n)];
      else if (n == m + 1) bd = 0.0f;
      else                 bd = BD[(size_t)(m + 1) * kS + (n - m - 2)];
      SC[(size_t)m * kS + n] = (acc[t][r] + bd) * scale;
    }
  }
}

// ---------------------------------------------------------------------------
// Row softmax in place over SC rows of length S. One 256-thread block per row.
// ---------------------------------------------------------------------------
__global__ void softmax_rows(float* __restrict__ SC) {
  __shared__ float red[256];
  const int tid = threadIdx.x;
  float* p = SC + (size_t)blockIdx.x * kS;

  float m = -INFINITY;
  for (int i = tid; i < kS; i += 256) m = fmaxf(m, p[i]);
  red[tid] = m; __syncthreads();
  for (int s = 128; s > 0; s >>= 1) {
    if (tid < s) red[tid] = fmaxf(red[tid], red[tid + s]);
    __syncthreads();
  }
  m = red[0]; __syncthreads();

  float sum = 0.0f;
  for (int i = tid; i < kS; i += 256) {
    float e = __expf(p[i] - m);
    p[i] = e;
    sum += e;
  }
  red[tid] = sum; __syncthreads();
  for (int s = 128; s > 0; s >>= 1) {
    if (tid < s) red[tid] += red[tid + s];
    __syncthreads();
  }
  float inv = 1.0f / red[0];
  for (int i = tid; i < kS; i += 256) p[i] *= inv;
}

// ---------------------------------------------------------------------------
// Per head: AVec[q, h*64+d] = sum_k P[q,k] * V[h][k][d]   (NN GEMM, K = 2048)
// Block = 8 waves, each owns one 16-row tile (all 64 output cols).
// ---------------------------------------------------------------------------
__global__ void av_gemm(const float* __restrict__ SC, const float* __restrict__ Vh,
                        float* __restrict__ AVec, int h) {
  const int lane = threadIdx.x & 31;
  const int wave = threadIdx.x >> 5;
  const int half = lane >> 4;
  const int lr   = lane & 15;
  const int m0   = (blockIdx.x * 8 + wave) * 16;

  const float* pp = SC + (size_t)(m0 + lr) * kS + 2 * half;
  const float* vp = Vh + (size_t)h * kS * kDH + lr;

  v8f acc[4] = {v8f{}, v8f{}, v8f{}, v8f{}};
  for (int k = 0; k < kS; k += 4) {
    v2f a = *(const v2f*)(pp + k);
    int kk = k + 2 * half;
#pragma unroll
    for (int t = 0; t < 4; ++t) {
      const float* vb = vp + t * 16;
      v2f b = {vb[(size_t)kk * kDH], vb[(size_t)(kk + 1) * kDH]};
      acc[t] = wmma_f32(a, b, acc[t]);
    }
  }
#pragma unroll
  for (int t = 0; t < 4; ++t) {
    int d = t * 16 + lr;
#pragma unroll
    for (int r = 0; r < 8; ++r) {
      int m = m0 + r + 8 * half;
      AVec[(size_t)m * kD + h * kDH + d] = acc[t][r];
    }
  }
}

// ---------------------------------------------------------------------------
// out[m,n] = sum_k AVec[m,k] * Wo[n,k]   (plain NT GEMM into d_out)
// ---------------------------------------------------------------------------
__global__ void out_gemm(const float* __restrict__ A, const float* __restrict__ B,
                         float* __restrict__ C) {
  const int lane = threadIdx.x & 31;
  const int wave = threadIdx.x >> 5;
  const int half = lane >> 4;
  const int lr   = lane & 15;
  const int m0   = blockIdx.x * 16;
  const int n0   = wave * 64;

  const float* ap = A + (size_t)(m0 + lr) * kD + 2 * half;
  const float* bp = B + (size_t)(n0 + lr) * kD + 2 * half;

  v8f acc[4] = {v8f{}, v8f{}, v8f{}, v8f{}};
  for (int k = 0; k < kD; k += 4) {
    v2f a = *(const v2f*)(ap + k);
#pragma unroll
    for (int t = 0; t < 4; ++t) {
      v2f b = *(const v2f*)(bp + (size_t)t * 16 * kD + k);
      acc[t] = wmma_f32(a, b, acc[t]);
    }
  }
#pragma unroll
  for (int t = 0; t < 4; ++t) {
    int n = n0 + t * 16 + lr;
#pragma unroll
    for (int r = 0; r < 8; ++r) {
      int m = m0 + r + 8 * half;
      C[(size_t)m * kD + n] = acc[t][r];
    }
  }
}

// ---------------------------------------------------------------------------
extern "C" void kernel_launch(void* const* d_in, const int* in_sizes, int n_in,
                              void* d_out, int out_size, void* d_ws, size_t ws_size,
                              hipStream_t stream) {
  const float* x   = (const float*)d_in[0];
  const float* pos = (const float*)d_in[1];
  const float* Wq  = (const float*)d_in[2];
  const float* Wk  = (const float*)d_in[3];
  const float* Wv  = (const float*)d_in[4];
  const float* Wp  = (const float*)d_in[5];
  const float* Wo  = (const float*)d_in[6];
  const float* u   = (const float*)d_in[7];
  const float* v   = (const float*)d_in[8];

  float* ws = (float*)d_ws;
  const size_t SD = (size_t)kS * kD;   // 1 Mi floats
  float* Qhs  = ws;                    // [H][S][DH]
  float* Khs  = Qhs + SD;
  float* Vhs  = Khs + SD;
  float* Phs  = Vhs + SD;
  float* AVec = Phs + SD;              // [S][D]
  float* BD   = AVec + SD;             // [S][S]  (per-head, reused)
  float* SC   = BD + (size_t)kS * kS;  // [S][S]  (per-head, reused)

  dim3 blk(256);
  // projections: x@Wq^T, x@Wk^T, x@Wv^T, pos@Wp^T -> head-major layouts
  proj_gemm_head<<<dim3(kS / 16), blk, 0, stream>>>(x,   Wq, Qhs);
  proj_gemm_head<<<dim3(kS / 16), blk, 0, stream>>>(x,   Wk, Khs);
  proj_gemm_head<<<dim3(kS / 16), blk, 0, stream>>>(x,   Wv, Vhs);
  proj_gemm_head<<<dim3(kS / 16), blk, 0, stream>>>(pos, Wp, Phs);

  const float scale = 0.125f;  // 1/sqrt(64)
  for (int h = 0; h < kH; ++h) {
    bd_gemm   <<<dim3(kS / 16, kS / 512), blk, 0, stream>>>(Qhs, Phs, v, BD, h);
    score_gemm<<<dim3(kS / 16, kS / 512), blk, 0, stream>>>(Qhs, Khs, u, BD, SC, h, scale);
    softmax_rows<<<dim3(kS), blk, 0, stream>>>(SC);
    av_gemm   <<<dim3(kS / 128), blk, 0, stream>>>(SC, Vhs, AVec, h);
  }

  out_gemm<<<dim3(kS / 16), blk, 0, stream>>>(AVec, Wo, (float*)d_out);
}